// energy_bc_loss_61375082659857
// MI455X (gfx1250) — compile-verified
//
#include <hip/hip_runtime.h>

typedef float v2f __attribute__((ext_vector_type(2)));
typedef float v8f __attribute__((ext_vector_type(8)));

constexpr int   B      = 4;
constexpr int   C      = 3;
constexpr int   H      = 256;
constexpr int   W      = 256;
constexpr int   HW     = H * W;
constexpr int   HP     = H - 2;          // 254
constexpr int   WP     = W - 2;          // 254
constexpr int   NPATCH = HP * WP;        // 64516
constexpr int   NSEL   = 65;             // int(0.001 * 65536)
constexpr float EPSV   = 1e-5f;
constexpr int   PAD    = 7;              // 15x15 window

// ---------------------------------------------------------------------------
// Async global->LDS copy (gfx1250 ASYNCcnt path, VGLOBAL encoding):
//   VDST = per-lane LDS byte offset, VADDR = 64-bit global address.
// Generic LDS pointer -> LDS offset by keeping the low 32 address bits.
// ---------------------------------------------------------------------------
__device__ __forceinline__ void async_load_f32(const float* gp, unsigned lds_off) {
    asm volatile("global_load_async_to_lds_b32 %0, %1, off"
                 :: "v"(lds_off), "v"(gp)
                 : "memory");
}
__device__ __forceinline__ void wait_async_all() {
    asm volatile("s_wait_asynccnt 0" ::: "memory");
}

// ---------------------------------------------------------------------------
// Wave-level sum reduction through the fp32 tensor pipe:
// A (16x4) holds the 32 lane partials, B (4x16) = ones, so D[i][j] = rowsum_i.
// Lane L<16 sums rows 0..7 of its column, lane L>=16 rows 8..15; one xor-16
// shuffle closes the wave. EXEC must be all ones at the call site.
// ---------------------------------------------------------------------------
__device__ __forceinline__ float wave_wmma_sum(float v) {
    v2f a;    a.x = v;    a.y = 0.0f;
    v2f ones; ones.x = 1.0f; ones.y = 1.0f;
    v8f acc = {};
    acc = __builtin_amdgcn_wmma_f32_16x16x4_f32(
        /*neg_a=*/false, a, /*neg_b=*/false, ones,
        /*c_mod=*/(short)0, acc, /*reuse_a=*/false, /*reuse_b=*/false);
    float t = 0.0f;
#pragma unroll
    for (int i = 0; i < 8; ++i) t += acc[i];
    t += __shfl_xor(t, 16);
    return t;   // full 32-lane sum, identical in every lane
}

// ---------------------------------------------------------------------------
__global__ void k_init(float* __restrict__ part) {
    if (threadIdx.x < 2) part[threadIdx.x] = 0.0f;
}

// channel-max then horizontal 15-max (row cached in LDS)
__global__ void k_cmax_rowmax(const float* __restrict__ img, float* __restrict__ tmp) {
    const int b = blockIdx.x >> 8;
    const int y = blockIdx.x & 255;
    const int x = threadIdx.x;
    __shared__ float row[W];
    const float* p = img + ((size_t)b * C * H + y) * W;
    float m = p[x];
    m = fmaxf(m, p[(size_t)HW + x]);
    m = fmaxf(m, p[(size_t)2 * HW + x]);
    row[x] = m;
    __syncthreads();
    float r = -1e30f;
#pragma unroll
    for (int dx = -PAD; dx <= PAD; ++dx) {
        int xx = x + dx;
        if (xx >= 0 && xx < W) r = fmaxf(r, row[xx]);
    }
    tmp[((size_t)b * H + y) * W + x] = r;
}

// vertical 15-max; optionally emit 1 - max (for t_slide)
__global__ void k_colmax(const float* __restrict__ tmp, float* __restrict__ out, int one_minus) {
    const int b = blockIdx.x >> 8;
    const int y = blockIdx.x & 255;
    const int x = threadIdx.x;
    float r = -1e30f;
#pragma unroll
    for (int dy = -PAD; dy <= PAD; ++dy) {
        int yy = y + dy;
        if (yy >= 0 && yy < H) r = fmaxf(r, tmp[((size_t)b * H + yy) * W + x]);
    }
    out[((size_t)b * H + y) * W + x] = one_minus ? (1.0f - r) : r;
}

// per batch: indices of the NSEL smallest bc values (bit-pattern binary search;
// bc >= 0 so uint compare is order-preserving). LDS atomics for counting.
__global__ void k_select(const float* __restrict__ bc, int* __restrict__ selIdx) {
    const int b   = blockIdx.x;
    const int tid = threadIdx.x;
    const float* p = bc + (size_t)b * HW;
    __shared__ unsigned s_cnt, s_lo, s_hi, s_lt, s_eq;
    if (tid == 0) { s_lo = 0u; s_hi = 0x3F800001u; }   // covers [0, 1.0] inclusive
    __syncthreads();
    for (int it = 0; it < 32; ++it) {
        unsigned lo = s_lo, hi = s_hi;
        if (lo >= hi) break;                            // uniform condition
        if (tid == 0) s_cnt = 0u;
        __syncthreads();
        unsigned mid = lo + ((hi - lo) >> 1);
        unsigned c = 0u;
        for (int i = tid; i < HW; i += 256)
            c += (__float_as_uint(p[i]) <= mid) ? 1u : 0u;
        atomicAdd(&s_cnt, c);
        __syncthreads();
        if (tid == 0) { if (s_cnt >= (unsigned)NSEL) s_hi = mid; else s_lo = mid + 1u; }
        __syncthreads();
    }
    const unsigned T = s_lo;                            // rank-NSEL threshold bits
    if (tid == 0) { s_lt = 0u; s_eq = 0u; }
    __syncthreads();
    unsigned clt = 0u;
    for (int i = tid; i < HW; i += 256)
        clt += (__float_as_uint(p[i]) < T) ? 1u : 0u;
    atomicAdd(&s_lt, clt);
    __syncthreads();
    const unsigned nlt = s_lt;
    if (tid == 0) s_cnt = 0u;
    __syncthreads();
    for (int i = tid; i < HW; i += 256) {
        unsigned v = __float_as_uint(p[i]);
        if (v < T) {
            unsigned pos = atomicAdd(&s_cnt, 1u);
            selIdx[b * NSEL + pos] = i;
        } else if (v == T) {
            unsigned pos = atomicAdd(&s_eq, 1u);
            if (nlt + pos < (unsigned)NSEL) selIdx[b * NSEL + nlt + pos] = i;
        }
    }
}

// atmosphere: mean of img over ALL B*NSEL selected pixel positions
// (faithfully reproduces the reference's idx.reshape(-1) cross-batch gather)
__global__ void k_atmo(const float* __restrict__ img, const int* __restrict__ selIdx,
                       float* __restrict__ A) {
    const int b   = blockIdx.x;
    const int tid = threadIdx.x;
    __shared__ float s0[256], s1[256], s2[256];
    float a0 = 0.f, a1 = 0.f, a2 = 0.f;
    const float* p = img + (size_t)b * C * HW;
    for (int j = tid; j < B * NSEL; j += 256) {
        int idx = selIdx[j];
        a0 += p[idx];
        a1 += p[(size_t)HW + idx];
        a2 += p[(size_t)2 * HW + idx];
    }
    s0[tid] = a0; s1[tid] = a1; s2[tid] = a2;
    __syncthreads();
    for (int off = 128; off > 0; off >>= 1) {
        if (tid < off) { s0[tid] += s0[tid + off]; s1[tid] += s1[tid + off]; s2[tid] += s2[tid + off]; }
        __syncthreads();
    }
    if (tid == 0) {
        const float inv = 1.0f / (float)(B * NSEL);
        A[b * 3 + 0] = s0[0] * inv;
        A[b * 3 + 1] = s1[0] * inv;
        A[b * 3 + 2] = s2[0] * inv;
    }
}

// channel-max of (1-I)/(1-A+eps) then horizontal 15-max
__global__ void k_norm_rowmax(const float* __restrict__ img, const float* __restrict__ A,
                              float* __restrict__ tmp) {
    const int b = blockIdx.x >> 8;
    const int y = blockIdx.x & 255;
    const int x = threadIdx.x;
    __shared__ float row[W];
    const float i0 = 1.0f / (1.0f - A[b * 3 + 0] + EPSV);
    const float i1 = 1.0f / (1.0f - A[b * 3 + 1] + EPSV);
    const float i2 = 1.0f / (1.0f - A[b * 3 + 2] + EPSV);
    const float* p = img + ((size_t)b * C * H + y) * W;
    float m = (1.0f - p[x]) * i0;
    m = fmaxf(m, (1.0f - p[(size_t)HW + x]) * i1);
    m = fmaxf(m, (1.0f - p[(size_t)2 * HW + x]) * i2);
    row[x] = m;
    __syncthreads();
    float r = -1e30f;
#pragma unroll
    for (int dx = -PAD; dx <= PAD; ++dx) {
        int xx = x + dx;
        if (xx >= 0 && xx < W) r = fmaxf(r, row[xx]);
    }
    tmp[((size_t)b * H + y) * W + x] = r;
}

// smoothness term: 16x16 patch tile per block, 18x18x(3+1) fp32 tile staged
// via async global->LDS DMA, closed-form symmetric 3x3 inverse, quadratic
// form via s^T F s, WMMA fp32 wave reduction, one global atomic per block.
constexpr int TS = 16;
constexpr int TL = TS + 2;   // 18
__global__ void k_smooth(const float* __restrict__ img, const float* __restrict__ yp,
                         float* __restrict__ part) {
    __shared__ float tile[4][TL * TL];
    __shared__ float s_ws[8];
    const int bpp = (WP + TS - 1) / TS;          // 16
    const int b   = blockIdx.x / (bpp * bpp);
    const int t   = blockIdx.x % (bpp * bpp);
    const int ty0 = (t / bpp) * TS;
    const int tx0 = (t % bpp) * TS;
    const int tid = threadIdx.x;
    const float* pi = img + (size_t)b * C * HW;
    const float* py = yp + (size_t)b * HW;
    // stage the tile with async global->LDS transfers (ASYNCcnt path)
    for (int e = tid; e < TL * TL; e += 256) {
        int ly = e / TL, lx = e % TL;
        int gy = min(ty0 + ly, H - 1);
        int gx = min(tx0 + lx, W - 1);
        size_t o = (size_t)gy * W + gx;
        async_load_f32(pi + o,                  (unsigned)(size_t)&tile[0][e]);
        async_load_f32(pi + (size_t)HW + o,     (unsigned)(size_t)&tile[1][e]);
        async_load_f32(pi + (size_t)2 * HW + o, (unsigned)(size_t)&tile[2][e]);
        async_load_f32(py + o,                  (unsigned)(size_t)&tile[3][e]);
    }
    wait_async_all();      // own wave's async transfers landed in LDS
    __syncthreads();       // cross-wave visibility
    const int ly = tid >> 4, lx = tid & 15;
    const int y = ty0 + ly, x = tx0 + lx;
    float contrib = 0.0f;
    if (y < HP && x < WP) {
        float S0 = 0.f, S1 = 0.f, S2 = 0.f;
        float P00 = 0.f, P01 = 0.f, P02 = 0.f, P11 = 0.f, P12 = 0.f, P22 = 0.f;
        float s1 = 0.f, s2 = 0.f;
        float X0[9], X1[9], X2[9];
#pragma unroll
        for (int dy = 0; dy < 3; ++dy)
#pragma unroll
            for (int dx = 0; dx < 3; ++dx) {
                int e = (ly + dy) * TL + (lx + dx);
                float a = tile[0][e], bb = tile[1][e], c = tile[2][e], yy = tile[3][e];
                int k = dy * 3 + dx;
                X0[k] = a; X1[k] = bb; X2[k] = c;
                S0 += a; S1 += bb; S2 += c;
                P00 += a * a;  P01 += a * bb; P02 += a * c;
                P11 += bb * bb; P12 += bb * c; P22 += c * c;
                s1 += yy; s2 += yy * yy;
            }
        const float inv9 = 1.0f / 9.0f;
        float u0 = S0 * inv9, u1 = S1 * inv9, u2 = S2 * inv9;
        float m00 = P00 * inv9 - u0 * u0 + EPSV * inv9;
        float m01 = P01 * inv9 - u0 * u1;
        float m02 = P02 * inv9 - u0 * u2;
        float m11 = P11 * inv9 - u1 * u1 + EPSV * inv9;
        float m12 = P12 * inv9 - u1 * u2;
        float m22 = P22 * inv9 - u2 * u2 + EPSV * inv9;
        float c00 = m11 * m22 - m12 * m12;
        float c01 = m02 * m12 - m01 * m22;
        float c02 = m01 * m12 - m02 * m11;
        float det = m00 * c00 + m01 * c01 + m02 * c02;
        float id  = 1.0f / det;
        float f00 = c00 * id, f01 = c01 * id, f02 = c02 * id;
        float f11 = (m00 * m22 - m02 * m02) * id;
        float f12 = (m01 * m02 - m00 * m12) * id;
        float f22 = (m00 * m11 - m01 * m01) * id;
        float t0 = 0.f, t1 = 0.f, t2 = 0.f;
#pragma unroll
        for (int k = 0; k < 9; ++k) { t0 += X0[k] - u0; t1 += X1[k] - u1; t2 += X2[k] - u2; }
        float Q = f00 * t0 * t0 + f11 * t1 * t1 + f22 * t2 * t2
                + 2.0f * (f01 * t0 * t1 + f02 * t0 * t2 + f12 * t1 * t2);
        float w_sum = (Q + 81.0f) * inv9;
        float nd = 18.0f * s2 - 2.0f * s1 * s1;
        contrib = nd * w_sum;
    }
    float tot = wave_wmma_sum(contrib);          // unconditional: EXEC all-1s
    if ((tid & 31) == 0) s_ws[tid >> 5] = tot;
    __syncthreads();
    if (tid == 0) {
        float s = 0.0f;
#pragma unroll
        for (int wv = 0; wv < 8; ++wv) s += s_ws[wv];
        atomicAdd(&part[0], s);
    }
}

// data term: sum (y_pred - t_slide)^2, float4 streaming, WMMA wave reduction
__global__ void k_data(const float* __restrict__ yp, const float* __restrict__ ts,
                       float* __restrict__ part) {
    __shared__ float s_ws[8];
    const float4* yp4 = (const float4*)yp;
    const float4* ts4 = (const float4*)ts;
    const int total4 = (B * HW) / 4;             // 65536
    const int stride = gridDim.x * blockDim.x;
    float acc = 0.0f;
    for (int i = blockIdx.x * blockDim.x + threadIdx.x; i < total4; i += stride) {
        if (i + stride < total4) __builtin_prefetch(&yp4[i + stride], 0, 1);
        float4 a = yp4[i];
        float4 c = ts4[i];
        float d0 = a.x - c.x, d1 = a.y - c.y, d2 = a.z - c.z, d3 = a.w - c.w;
        acc += d0 * d0 + d1 * d1 + d2 * d2 + d3 * d3;
    }
    float tot = wave_wmma_sum(acc);              // unconditional: EXEC all-1s
    if ((threadIdx.x & 31) == 0) s_ws[threadIdx.x >> 5] = tot;
    __syncthreads();
    if (threadIdx.x == 0) {
        float s = 0.0f;
#pragma unroll
        for (int wv = 0; wv < 8; ++wv) s += s_ws[wv];
        atomicAdd(&part[1], s);
    }
}

__global__ void k_final(const float* __restrict__ part, float* __restrict__ out) {
    if (threadIdx.x == 0 && blockIdx.x == 0)
        out[0] = (part[0] + 0.001f * part[1]) / (float)NPATCH;
}

// ---------------------------------------------------------------------------
extern "C" void kernel_launch(void* const* d_in, const int* in_sizes, int n_in,
                              void* d_out, int out_size, void* d_ws, size_t ws_size,
                              hipStream_t stream) {
    const float* img = (const float*)d_in[0];   // (4,3,256,256) f32
    const float* yp  = (const float*)d_in[1];   // (4,1,256,256) f32
    float* ws   = (float*)d_ws;
    float* buf0 = ws;                                 // B*HW: row-max scratch
    float* buf1 = ws + (size_t)B * HW;                // B*HW: bc, then t_slide
    float* A    = ws + (size_t)2 * B * HW;            // 16 floats (B*3 padded)
    int*   sel  = (int*)(ws + (size_t)2 * B * HW + 16);      // B*NSEL = 260 ints
    float* part = ws + (size_t)2 * B * HW + 16 + 272;        // 2 accumulators
    float* out  = (float*)d_out;

    k_init<<<1, 32, 0, stream>>>(part);
    k_cmax_rowmax<<<B * H, W, 0, stream>>>(img, buf0);
    k_colmax<<<B * H, W, 0, stream>>>(buf0, buf1, 0);        // buf1 = bright channel
    k_select<<<B, 256, 0, stream>>>(buf1, sel);
    k_atmo<<<B, 256, 0, stream>>>(img, sel, A);
    k_norm_rowmax<<<B * H, W, 0, stream>>>(img, A, buf0);
    k_colmax<<<B * H, W, 0, stream>>>(buf0, buf1, 1);        // buf1 = t_slide
    k_smooth<<<B * 16 * 16, 256, 0, stream>>>(img, yp, part);
    k_data<<<256, 256, 0, stream>>>(yp, buf1, part);
    k_final<<<1, 32, 0, stream>>>(part, out);
}